// YOLODetection_87093346828915
// MI455X (gfx1250) — compile-verified
//
#include <hip/hip_runtime.h>
#include <cstdint>

#define GDIM   52
#define ADIM   3
#define BDIM   32
#define CCH    82          // N_TOP + N_BOTTOM
#define NCH    88          // 5 + 1 + CCH
#define PLANE  (GDIM*GDIM) // 2704
#define NCELLS (BDIM*ADIM*PLANE) // 259584
#define NTOP   53
#define STRIDEF 8.0f       // 416/52
#define IMGSZ  416.0f
#define NO_OBJ_W 10.0f

// workspace layout (floats):
//   [0..15]      accumulators
//   [16..1039]   1024 block partials (padded with zeros for WMMA reduction)
//   [1040.. ]    hit flags, 1 byte per cell (259584 bytes)
#define ACC_MSE       0
#define ACC_CONF_OBJ  1
#define ACC_ISTOP     2
#define ACC_CLS_TOP   3
#define ACC_CLS_BOT   4
#define ACC_NTOPBOX   5
#define ACC_HITCELLS  6
#define PART_OFF      16
#define NPART         1024
#define HIT_OFF_F     1040

typedef __attribute__((ext_vector_type(2))) float v2f;
typedef __attribute__((ext_vector_type(8))) float v8f;

__device__ __forceinline__ float sigm(float z) { return 1.0f / (1.0f + expf(-z)); }

// min(softplus(z), 100) == -clip(log(sigmoid(-z)), -100)
__device__ __forceinline__ float splus100(float z) {
    float sp = (z > 20.0f) ? z : log1pf(expf(z));
    return fminf(sp, 100.0f);
}
// -(t*clip(log p,-100) + (1-t)*clip(log1p(-p),-100)),  p = sigmoid(z)
__device__ __forceinline__ float bce_term(float z, float t) {
    return t * splus100(-z) + (1.0f - t) * splus100(z);
}
__device__ __forceinline__ float mmean(float s, float cnt) {
    return (cnt > 0.5f) ? s / cnt : 0.0f;
}

__global__ void k_init(float* wsf) {
    int t = blockIdx.x * 256 + threadIdx.x;
    uint32_t* hit32 = (uint32_t*)(wsf + HIT_OFF_F);
    if (t < NCELLS / 4) hit32[t] = 0u;
    if (t < NPART) wsf[PART_OFF + t] = 0.0f;
    if (t < 16) wsf[t] = 0.0f;
}

// One thread per target box; all (bi,cj,ci) are distinct by construction,
// so hit-flag writes and per-cell loss terms are conflict-free.
__global__ void k_boxes(const float* __restrict__ x, const float* __restrict__ tgt,
                        const float* __restrict__ anc, float* wsf, int N) {
    __shared__ float sred[512];
    int n = threadIdx.x;
    float vals[7] = {0.f, 0.f, 0.f, 0.f, 0.f, 0.f, 0.f};
    uint8_t* hit = (uint8_t*)(wsf + HIT_OFF_F);

    if (n < N) {
        const float* t = tgt + (size_t)n * NCH;
        int bi = (int)t[0];
        float tcx = t[1] * (float)GDIM, tcy = t[2] * (float)GDIM;
        float tw = t[3], th = t[4], istop = t[5];
        int ci = (int)tcx, cj = (int)tcy;

        float iou[3];
        #pragma unroll
        for (int a = 0; a < 3; a++) {
            float aw = anc[2*a], ah = anc[2*a+1];
            float inter = fminf(aw, tw) * fminf(ah, th);
            float uni   = tw * th + aw * ah - inter + 1e-16f;
            iou[a] = inter / uni;
        }
        int best = 0;                       // first-max semantics like jnp.argmax
        if (iou[1] > iou[best]) best = 1;
        if (iou[2] > iou[best]) best = 2;

        int cellbase = (bi * ADIM) * PLANE + cj * GDIM + ci;
        float hits = 0.0f;
        #pragma unroll
        for (int a = 0; a < 3; a++) {
            bool f = (iou[a] > 0.5f) || (a == best);
            if (f) { hit[cellbase + a * PLANE] = 1; hits += 1.0f; }
        }
        vals[ACC_HITCELLS] = hits;

        const float* xp = x + (size_t)(bi * ADIM + best) * NCH * PLANE
                            + (size_t)cj * GDIM + ci;
        float z0 = xp[0],          z1 = xp[PLANE],     z2 = xp[2*PLANE];
        float z3 = xp[3*PLANE],    z4 = xp[4*PLANE],   z5 = xp[5*PLANE];
        float aw = anc[2*best], ah = anc[2*best+1];

        float d0 = sigm(z0) - (tcx - floorf(tcx));
        float d1 = sigm(z1) - (tcy - floorf(tcy));
        float d2 = z2 - logf(tw / aw + 1e-16f);
        float d3 = z3 - logf(th / ah + 1e-16f);
        vals[ACC_MSE]      = d0*d0 + d1*d1 + d2*d2 + d3*d3;
        vals[ACC_CONF_OBJ] = splus100(-z4);           // bce with t=1
        vals[ACC_ISTOP]    = bce_term(z5, istop);

        if (istop > 0.5f) {
            float s = 0.0f;
            for (int c = 0; c < NTOP; c++)
                s += bce_term(xp[(size_t)(6 + c) * PLANE], t[6 + c]);
            vals[ACC_CLS_TOP] = s;
            vals[ACC_NTOPBOX] = 1.0f;
        } else {
            float s = 0.0f;
            for (int c = NTOP; c < CCH; c++)
                s += bce_term(xp[(size_t)(6 + c) * PLANE], t[6 + c]);
            vals[ACC_CLS_BOT] = s;
        }
    }

    // deterministic tree reductions, one accumulator at a time
    for (int q = 0; q < 7; q++) {
        sred[threadIdx.x] = vals[q];
        __syncthreads();
        for (int off = 256; off > 0; off >>= 1) {
            if ((int)threadIdx.x < off) sred[threadIdx.x] += sred[threadIdx.x + off];
            __syncthreads();
        }
        if (threadIdx.x == 0) wsf[q] = sred[0];
        __syncthreads();
    }
}

// One thread per cell: elementwise output transform + per-block no-obj BCE partial.
// Loads: per channel plane, lanes walk contiguous (j*52+i) -> coalesced.
// Stores: 22 x float4 per cell (88 contiguous floats), block covers region densely.
__global__ void k_stream(const float* __restrict__ x, const float* __restrict__ anc,
                         float* __restrict__ out, float* wsf) {
    __shared__ float sred[256];
    int gid   = blockIdx.x * 256 + threadIdx.x;          // < 259584 exactly
    int plane = gid / PLANE;
    int s     = gid - plane * PLANE;
    int a     = plane % ADIM;
    int j     = s / GDIM;
    int i     = s - j * GDIM;

    const float* xin = x + (size_t)plane * NCH * PLANE + s;
    float*       op  = out + (size_t)gid * NCH;
    const uint8_t* hit = (const uint8_t*)(wsf + HIT_OFF_F);

    float aw = anc[2*a] * IMGSZ, ah = anc[2*a+1] * IMGSZ;
    float fi = (float)i, fj = (float)j;
    int   h  = hit[gid];
    float local = 0.0f;

    #pragma unroll
    for (int k = 0; k < 22; k++) {
        float zz[4];
        #pragma unroll
        for (int e = 0; e < 4; e++)
            zz[e] = xin[(size_t)(4*k + e) * PLANE];
        #pragma unroll
        for (int e = 0; e < 4; e++) {
            int ch = 4*k + e;           // literal after unroll -> branches fold
            float z = zz[e], r;
            if      (ch == 0) r = (fi + sigm(z)) * STRIDEF;
            else if (ch == 1) r = (fj + sigm(z)) * STRIDEF;
            else if (ch == 2) r = aw * expf(z);
            else if (ch == 3) r = ah * expf(z);
            else if (ch == 4) { r = sigm(z); if (h == 0) local += splus100(z); }
            else              r = sigm(z);
            zz[e] = r;
        }
        float4 w4 = make_float4(zz[0], zz[1], zz[2], zz[3]);
        *(float4*)(op + 4*k) = w4;
    }

    sred[threadIdx.x] = local;
    __syncthreads();
    for (int off = 128; off > 0; off >>= 1) {
        if ((int)threadIdx.x < off) sred[threadIdx.x] += sred[threadIdx.x + off];
        __syncthreads();
    }
    if (threadIdx.x == 0) wsf[PART_OFF + blockIdx.x] = sred[0];
}

// Single wave: exact reduction of 1024 partials via 16 chained
// V_WMMA_F32_16X16X4_F32 with B = ones (D = A*1 + C accumulates row sums),
// then compose the final scalar loss.
__global__ void k_final(float* wsf, float* out, int lossIdx, int N) {
    int lane = threadIdx.x;
    const float* p = wsf + PART_OFF;

    v8f c = {0.f, 0.f, 0.f, 0.f, 0.f, 0.f, 0.f, 0.f};
    v2f bone = {1.0f, 1.0f};
    // A(16x4) layout: lanes 0-15 hold (M=lane, K=0/1); lanes 16-31 (M=lane-16, K=2/3).
    // Map A[m,k] = partial[chunk*64 + 4m + k].
    int idx0 = 4 * (lane & 15) + 2 * (lane >> 4);
    for (int chunk = 0; chunk < 16; ++chunk) {
        v2f av;
        av.x = p[chunk * 64 + idx0];
        av.y = p[chunk * 64 + idx0 + 1];
        c = __builtin_amdgcn_wmma_f32_16x16x4_f32(false, av, false, bone,
                                                  (short)0, c, false, false);
    }
    // Column 0 of D lives in lane 0 (M=0..7) and lane 16 (M=8..15), 8 VGPRs each.
    float srow = c[0]+c[1]+c[2]+c[3]+c[4]+c[5]+c[6]+c[7];
    float noobj_sum = __shfl(srow, 0, 32) + __shfl(srow, 16, 32);

    if (lane == 0) {
        float mse    = wsf[ACC_MSE];
        float cobjs  = wsf[ACC_CONF_OBJ];
        float istops = wsf[ACC_ISTOP];
        float clstop = wsf[ACC_CLS_TOP];
        float clsbot = wsf[ACC_CLS_BOT];
        float ntop   = wsf[ACC_NTOPBOX];
        float hitc   = wsf[ACC_HITCELLS];

        float cobj = (float)N;                       // all target cells distinct
        float cno  = (float)NCELLS - hitc;
        float ctop = ntop * (float)NTOP;
        float cbot = ((float)N - ntop) * (float)(CCH - NTOP);

        float loss = mmean(mse, cobj)
                   + mmean(cobjs, cobj) + NO_OBJ_W * mmean(noobj_sum, cno)
                   + mmean(istops, cobj)
                   + mmean(clstop, ctop) + mmean(clsbot, cbot);
        out[lossIdx] = loss;
    }
}

extern "C" void kernel_launch(void* const* d_in, const int* in_sizes, int n_in,
                              void* d_out, int out_size, void* d_ws, size_t ws_size,
                              hipStream_t stream) {
    const float* x   = (const float*)d_in[0];
    const float* tgt = (const float*)d_in[1];
    const float* anc = (const float*)d_in[2];
    float* out = (float*)d_out;
    float* wsf = (float*)d_ws;
    int N = in_sizes[1] / NCH;   // 512

    k_init  <<<254, 256, 0, stream>>>(wsf);
    k_boxes <<<1, 512, 0, stream>>>(x, tgt, anc, wsf, N);
    k_stream<<<NCELLS / 256, 256, 0, stream>>>(x, anc, out, wsf);
    k_final <<<1, 32, 0, stream>>>(wsf, out, out_size - 1, N);
}